// DynamicScatter3D_1726576856007
// MI455X (gfx1250) — compile-verified
//
#include <hip/hip_runtime.h>

#define VGRID 64
#define NVOX  (VGRID * VGRID * VGRID)   // 262144 voxels
#define CHAN  64

typedef float    v4f __attribute__((ext_vector_type(4)));
typedef unsigned v4u __attribute__((ext_vector_type(4)));

// Monotonic map: float total order -> unsigned total order (bitwise-exact roundtrip).
__device__ __forceinline__ unsigned enc_f32(float f) {
    unsigned u = __float_as_uint(f);
    return u ^ (unsigned)(((int)u >> 31) | (int)0x80000000);
}
__device__ __forceinline__ float dec_f32(unsigned k) {
    unsigned u = k ^ (unsigned)((((int)(~k)) >> 31) | (int)0x80000000);
    return __uint_as_float(u);
}

// ---------------------------------------------------------------------------
// Kernel 1: init. Zero the ordered-key accumulation surface (key 0 == unsigned
// minimum == "no point seen"), zero float counts, write static voxel coords.
// ---------------------------------------------------------------------------
__global__ void k_init(unsigned* __restrict__ featsKeys,
                       float* __restrict__ coorsOut,
                       float* __restrict__ countOut) {
    unsigned t = blockIdx.x * blockDim.x + threadIdx.x;
    const unsigned totalVec = (unsigned)NVOX * CHAN / 4u;   // 4,194,304 uint4 stores
    if (t < totalVec) {
        v4u z = {0u, 0u, 0u, 0u};
        ((v4u*)featsKeys)[t] = z;                           // global_store_b128
    }
    if (t < (unsigned)NVOX) {
        unsigned x = t >> 12, y = (t >> 6) & 63u, z = t & 63u;
        coorsOut[3u * t + 0u] = (float)x;
        coorsOut[3u * t + 1u] = (float)y;
        coorsOut[3u * t + 2u] = (float)z;
        countOut[t] = 0.0f;
    }
}

// ---------------------------------------------------------------------------
// Kernel 2: scatter. 16 lanes per point; each lane streams 4 channels with a
// non-temporal b128 load (keep L2 free for the 65 MB atomic surface) and does
// 4 global_atomic_max_u32 on the ordered keys. Lane 0 counts the point.
// ---------------------------------------------------------------------------
__global__ void k_scatter(const float* __restrict__ points,
                          const int* __restrict__ coors,
                          unsigned* __restrict__ featsKeys,
                          float* __restrict__ countOut,
                          int npoints) {
    int t = blockIdx.x * blockDim.x + threadIdx.x;
    int point = t >> 4;
    int sub   = t & 15;
    if (point >= npoints) return;

    // Prefetch the points tile ~4096 points ahead (streaming hint).
    int pf = point + 4096;
    if (pf < npoints) {
        __builtin_prefetch(points + (size_t)pf * CHAN + sub * 4, 0, 0); // global_prefetch_b8
    }

    int cx = coors[3 * point + 0];
    int cy = coors[3 * point + 1];
    int cz = coors[3 * point + 2];
    int seg = (cx * VGRID + cy) * VGRID + cz;

    v4f p = __builtin_nontemporal_load(
        (const v4f*)(points + (size_t)point * CHAN) + sub);   // b128, TH=NT

    unsigned* dst = featsKeys + (size_t)seg * CHAN + (size_t)(sub * 4);
    atomicMax(dst + 0, enc_f32(p.x));
    atomicMax(dst + 1, enc_f32(p.y));
    atomicMax(dst + 2, enc_f32(p.z));
    atomicMax(dst + 3, enc_f32(p.w));

    if (sub == 0) {
        atomicAdd(countOut + seg, 1.0f);   // exact integer count in f32
    }
}

// ---------------------------------------------------------------------------
// Kernel 3: finalize. Decode ordered keys back to floats in place; rows with
// count==0 become zero (reference semantics for empty voxels).
// ---------------------------------------------------------------------------
__global__ void k_finalize(float* __restrict__ feats,          // keys in, floats out
                           const float* __restrict__ countOut) {
    unsigned t = blockIdx.x * blockDim.x + threadIdx.x;
    if (t >= (unsigned)NVOX * 16u) return;
    unsigned vox = t >> 4;
    float cnt = countOut[vox];
    v4u k = ((const v4u*)feats)[t];
    v4f o;
    if (cnt > 0.0f) {
        o.x = dec_f32(k.x); o.y = dec_f32(k.y);
        o.z = dec_f32(k.z); o.w = dec_f32(k.w);
    } else {
        o = (v4f){0.0f, 0.0f, 0.0f, 0.0f};
    }
    ((v4f*)feats)[t] = o;
}

// ---------------------------------------------------------------------------
extern "C" void kernel_launch(void* const* d_in, const int* in_sizes, int n_in,
                              void* d_out, int out_size, void* d_ws, size_t ws_size,
                              hipStream_t stream) {
    const float* points = (const float*)d_in[0];   // [N, 64] f32
    const int*   coors  = (const int*)d_in[1];     // [N, 3]  i32
    int npoints = in_sizes[1] / 3;

    float* out      = (float*)d_out;
    float* featsOut = out;                                   // [NVOX*64]
    float* coorsOut = out + (size_t)NVOX * CHAN;             // [NVOX*3]
    float* countOut = coorsOut + (size_t)NVOX * 3;           // [NVOX]
    unsigned* featsKeys = (unsigned*)featsOut;

    const int threads = 256;   // 8 wave32 per workgroup

    {
        int blocks = (NVOX * CHAN / 4 + threads - 1) / threads;   // 16384
        k_init<<<blocks, threads, 0, stream>>>(featsKeys, coorsOut, countOut);
    }
    {
        long total = (long)npoints * 16;
        int blocks = (int)((total + threads - 1) / threads);      // 62500
        k_scatter<<<blocks, threads, 0, stream>>>(points, coors, featsKeys,
                                                  countOut, npoints);
    }
    {
        int blocks = (NVOX * 16 + threads - 1) / threads;         // 16384
        k_finalize<<<blocks, threads, 0, stream>>>(featsOut, countOut);
    }
}